// InterContactLoss_18433999635091
// MI455X (gfx1250) — compile-verified
//
#include <hip/hip_runtime.h>
#include <math.h>

#define DIST_THR    0.2f
#define CONTACT_TGT 0.1f
#define INV_ALPHA   100.0f
#define HUBER_D     0.01f
#define EPS_F       1e-6f
#define TM          4          // A row-tiles (of 16 rows) per wave

typedef __attribute__((ext_vector_type(2))) float v2f;
typedef __attribute__((ext_vector_type(8))) float v8f;

// Build one B' column-tile in the ISA 4x16 B layout:
//   vgpr0 = K0 (bx) / K2 (bz) per half-wave, vgpr1 = K1 (by) / K3 (|b|^2).
// Padded columns get (0,0,0,1e30): big-finite so they never win the min and
// never create 0*INF NaNs.
__device__ __forceinline__ v2f load_btile(const float* __restrict__ Bb, int tb,
                                          int NB, int half, int l15) {
  const int n      = tb * 16 + l15;
  const bool valid = (n < NB);
  const int nc     = valid ? n : (NB - 1);
  const float bx = Bb[(size_t)nc * 3 + 0];
  const float by = Bb[(size_t)nc * 3 + 1];
  const float bz = Bb[(size_t)nc * 3 + 2];
  const float bb = bx * bx + by * by + bz * bz;
  float b0 = half ? bz : bx;
  float b1 = half ? bb : by;
  if (!valid) { b0 = 0.0f; b1 = half ? 1.0e30f : 0.0f; }
  v2f r; r.x = b0; r.y = b1;
  return r;
}

// One wave (32 lanes) per block computes min_n ||A[m]-B[n]|| for TM*16 rows m.
// S = A' x B' via V_WMMA_F32_16X16X4_F32 with
//   A'[m] = (-2ax, -2ay, -2az, 1),  B'[n] = (bx, by, bz, |b|^2)
// so S[m][n] = -2 a.b + |b|^2 and d2[m][n] = aa[m] + S[m][n].
// Inner loop: 2 B'-tiles x TM A'-tiles -> 8 WMMAs, folded into the running
// min with raw v_min3_num_f32 (no NaN-canonicalize overhead).
__global__ __launch_bounds__(32)
void dmin_wmma_kernel(const float* __restrict__ A, const float* __restrict__ B,
                      float* __restrict__ out, int NA, int NB) {
  const int lane = threadIdx.x;      // 0..31
  const int half = lane >> 4;        // 0 | 1
  const int l15  = lane & 15;
  const int bt    = blockIdx.y;
  const int tile0 = blockIdx.x * TM;

  const float* Ab = A + (size_t)bt * NA * 3;
  const float* Bb = B + (size_t)bt * NB * 3;

  // ---- A tiles (ISA 16x4 f32 layout: lanes0-15 K=0,1 ; lanes16-31 K=2,3) ----
  v2f   avec[TM];
  float aaAll[TM];                   // |a|^2 of row (tile t, row l15)
#pragma unroll
  for (int t = 0; t < TM; ++t) {
    const int m  = (tile0 + t) * 16 + l15;
    const int mc = (m < NA) ? m : (NA - 1);
    const float ax = Ab[(size_t)mc * 3 + 0];
    const float ay = Ab[(size_t)mc * 3 + 1];
    const float az = Ab[(size_t)mc * 3 + 2];
    aaAll[t] = ax * ax + ay * ay + az * az;
    avec[t].x = half ? (-2.0f * az) : (-2.0f * ax);   // K=2 : K=0
    avec[t].y = half ? 1.0f         : (-2.0f * ay);   // K=3 : K=1
  }

  float acc[TM][8];
#pragma unroll
  for (int t = 0; t < TM; ++t)
#pragma unroll
    for (int i = 0; i < 8; ++i) acc[t][i] = 3.0e38f;

  const int nTilesB = (NB + 15) >> 4;
  int tb = 0;
  for (; tb + 2 <= nTilesB; tb += 2) {
    const v2f b0v = load_btile(Bb, tb,     NB, half, l15);
    const v2f b1v = load_btile(Bb, tb + 1, NB, half, l15);
#pragma unroll
    for (int t = 0; t < TM; ++t) {
      v8f c = {0.f, 0.f, 0.f, 0.f, 0.f, 0.f, 0.f, 0.f};
      v8f d0 = __builtin_amdgcn_wmma_f32_16x16x4_f32(false, avec[t], false, b0v,
                                                     (short)0, c, false, false);
      v8f d1 = __builtin_amdgcn_wmma_f32_16x16x4_f32(false, avec[t], false, b1v,
                                                     (short)0, c, false, false);
#pragma unroll
      for (int i = 0; i < 8; ++i) {
        float e0 = d0[i], e1 = d1[i];
        asm("v_min3_num_f32 %0, %0, %1, %2"
            : "+v"(acc[t][i]) : "v"(e0), "v"(e1));
      }
    }
  }
  // remainder B-tile (at most one)
  for (; tb < nTilesB; ++tb) {
    const v2f bv = load_btile(Bb, tb, NB, half, l15);
#pragma unroll
    for (int t = 0; t < TM; ++t) {
      v8f c = {0.f, 0.f, 0.f, 0.f, 0.f, 0.f, 0.f, 0.f};
      v8f d = __builtin_amdgcn_wmma_f32_16x16x4_f32(false, avec[t], false, bv,
                                                    (short)0, c, false, false);
#pragma unroll
      for (int i = 0; i < 8; ++i) {
        float e = d[i];
        asm("v_min_num_f32 %0, %0, %1" : "+v"(acc[t][i]) : "v"(e));
      }
    }
  }

  // min over N: reduce across the 16 lanes of each half (wave32 xor shuffle)
#pragma unroll
  for (int t = 0; t < TM; ++t) {
#pragma unroll
    for (int i = 0; i < 8; ++i) {
      float v = acc[t][i];
      v = fminf(v, __shfl_xor(v, 1, 32));
      v = fminf(v, __shfl_xor(v, 2, 32));
      v = fminf(v, __shfl_xor(v, 4, 32));
      v = fminf(v, __shfl_xor(v, 8, 32));
      acc[t][i] = v;
    }
    // C layout: VGPR v -> tile-row v (half 0) / v+8 (half 1). Lane l15==0 stores.
#pragma unroll
    for (int v = 0; v < 8; ++v) {
      const float rowAA = __shfl(aaAll[t], v + 8 * half, 32);
      const float d2 = rowAA + acc[t][v];
      const float dv = sqrtf(fmaxf(d2, 0.0f));
      const int mOut = (tile0 + t) * 16 + v + 8 * half;
      if (l15 == 0 && mOut < NA) out[(size_t)bt * NA + mOut] = dv;
    }
  }
}

// Per-block partial sums: {pos_count, sum(-log p | t=1), sum(-log(1-p) | t=0), sum(huber*t)}
__global__ __launch_bounds__(256)
void stats_kernel(const float* __restrict__ dmin, const float* __restrict__ dmin_gt,
                  float* __restrict__ partials, int n) {
  __shared__ float sP[256], sLp[256], sLn[256], sH[256];
  float pos = 0.f, lp = 0.f, ln_ = 0.f, hb = 0.f;
  for (int i = blockIdx.x * blockDim.x + threadIdx.x; i < n;
       i += gridDim.x * blockDim.x) {
    const float d = dmin[i];
    const float t = (dmin_gt[i] < DIST_THR) ? 1.0f : 0.0f;
    const float x = (DIST_THR - d) * INV_ALPHA;
    float p = 1.0f / (1.0f + expf(-x));
    p = fminf(fmaxf(p, 1e-6f), 1.0f - 1e-6f);
    if (t > 0.5f) { pos += 1.0f; lp += -logf(p); }
    else          { ln_ += -logf(1.0f - p); }
    const float over = fmaxf(d - CONTACT_TGT, 0.0f);
    const float hub  = (over < HUBER_D) ? (0.5f * over * over / HUBER_D)
                                        : (over - 0.5f * HUBER_D);
    hb += t * hub;
  }
  const int tid = threadIdx.x;
  sP[tid] = pos; sLp[tid] = lp; sLn[tid] = ln_; sH[tid] = hb;
  __syncthreads();
  for (int s = 128; s > 0; s >>= 1) {
    if (tid < s) {
      sP[tid]  += sP[tid + s];
      sLp[tid] += sLp[tid + s];
      sLn[tid] += sLn[tid + s];
      sH[tid]  += sH[tid + s];
    }
    __syncthreads();
  }
  if (tid == 0) {
    partials[blockIdx.x * 4 + 0] = sP[0];
    partials[blockIdx.x * 4 + 1] = sLp[0];
    partials[blockIdx.x * 4 + 2] = sLn[0];
    partials[blockIdx.x * 4 + 3] = sH[0];
  }
}

__global__ void finalize_kernel(const float* __restrict__ ph, const float* __restrict__ po,
                                int nbH, int nbO, int Nh, int No, float* __restrict__ out) {
  if (threadIdx.x != 0 || blockIdx.x != 0) return;
  float aH[4] = {0.f, 0.f, 0.f, 0.f}, aO[4] = {0.f, 0.f, 0.f, 0.f};
  for (int i = 0; i < nbH; ++i)
    for (int k = 0; k < 4; ++k) aH[k] += ph[i * 4 + k];
  for (int i = 0; i < nbO; ++i)
    for (int k = 0; k < 4; ++k) aO[k] += po[i * 4 + k];

  const float negH = (float)Nh - aH[0];
  const float pwH  = (negH + EPS_F) / (aH[0] + EPS_F);
  const float bceH = (pwH * aH[1] + aH[2]) / (float)Nh;

  const float negO = (float)No - aO[0];
  const float pwO  = (negO + EPS_F) / (aO[0] + EPS_F);
  const float bceO = (pwO * aO[1] + aO[2]) / (float)No;

  const float hubH = (aH[0] > 0.f) ? aH[3] / fmaxf(aH[0], 1.f) : 0.f;
  const float hubO = (aO[0] > 0.f) ? aO[3] / fmaxf(aO[0], 1.f) : 0.f;

  // W_BCE * 0.5*(bceH+bceO) + W_DIST * (hubH+hubO)
  out[0] = 0.5f * (0.5f * (bceH + bceO)) + (hubH + hubO);
}

extern "C" void kernel_launch(void* const* d_in, const int* in_sizes, int n_in,
                              void* d_out, int out_size, void* d_ws, size_t ws_size,
                              hipStream_t stream) {
  const float* hv  = (const float*)d_in[0];
  const float* ov  = (const float*)d_in[1];
  const float* ghv = (const float*)d_in[2];
  const float* gov = (const float*)d_in[3];

  const int BT = 16;
  const int Vh = in_sizes[0] / (BT * 3);   // 6890
  const int Vo = in_sizes[1] / (BT * 3);   // 2048

  float* ws = (float*)d_ws;
  float* dmin_h  = ws;
  float* dmin_o  = dmin_h  + (size_t)BT * Vh;
  float* dmin_hg = dmin_o  + (size_t)BT * Vo;
  float* dmin_og = dmin_hg + (size_t)BT * Vh;
  float* partH   = dmin_og + (size_t)BT * Vo;
  const int RB = 64;
  float* partO   = partH + RB * 4;

  const int tilesH = (Vh + 15) / 16;
  const int tilesO = (Vo + 15) / 16;
  dim3 blk(32);
  dim3 gH((tilesH + TM - 1) / TM, BT);
  dim3 gO((tilesO + TM - 1) / TM, BT);
  dmin_wmma_kernel<<<gH, blk, 0, stream>>>(hv,  ov,  dmin_h,  Vh, Vo);
  dmin_wmma_kernel<<<gO, blk, 0, stream>>>(ov,  hv,  dmin_o,  Vo, Vh);
  dmin_wmma_kernel<<<gH, blk, 0, stream>>>(ghv, gov, dmin_hg, Vh, Vo);
  dmin_wmma_kernel<<<gO, blk, 0, stream>>>(gov, ghv, dmin_og, Vo, Vh);

  stats_kernel<<<RB, 256, 0, stream>>>(dmin_h, dmin_hg, partH, BT * Vh);
  stats_kernel<<<RB, 256, 0, stream>>>(dmin_o, dmin_og, partO, BT * Vo);
  finalize_kernel<<<1, 32, 0, stream>>>(partH, partO, RB, RB, BT * Vh, BT * Vo,
                                        (float*)d_out);
}